// TransformerLM_46385646797430
// MI455X (gfx1250) — compile-verified
//
#include <hip/hip_runtime.h>
#include <hip/hip_bf16.h>
#include <math.h>
#include <stdint.h>

#define V_   32000
#define L_   4
#define NH_  16
#define DM_  1024
#define DH_  64
#define DFF_ 4096
#define S_   1024
#define B_   2
#define MEM_ 1024
#define T_   (S_ + MEM_)   // 2048

typedef __attribute__((ext_vector_type(2))) float v2f;
typedef __attribute__((ext_vector_type(8))) float v8f;

// CDNA5 f32 WMMA: D(16x16,f32) = A(16x4,f32) * B(4x16,f32) + C
__device__ __forceinline__ v8f wmma4(v2f a, v2f b, v8f c) {
  return __builtin_amdgcn_wmma_f32_16x16x4_f32(false, a, false, b, (short)0, c,
                                               false, false);
}

// Wave-local LDS ordering (CDNA5 split counters)
__device__ __forceinline__ void lds_fence() {
  asm volatile("s_wait_dscnt 0" ::: "memory");
}

// CDNA5 async copy global -> LDS (tracked by ASYNCcnt). lds_off is the 32-bit
// LDS byte offset (low half of the flat shared address), g is the global addr.
__device__ __forceinline__ void async_b128(unsigned lds_off, const float* g) {
  asm volatile("global_load_async_to_lds_b128 %0, %1, off"
               :: "v"(lds_off), "v"(g) : "memory");
}
__device__ __forceinline__ void wait_async0() {
  asm volatile("s_wait_asynccnt 0" ::: "memory");
}

// ---------------- elementwise kernels ----------------

__global__ void embed_k(const int* __restrict__ inp, const float* __restrict__ emb,
                        float* __restrict__ h) {
  size_t idx = (size_t)blockIdx.x * blockDim.x + threadIdx.x;
  if (idx >= (size_t)S_ * B_ * DM_) return;
  int d  = (int)(idx & (DM_ - 1));
  int sb = (int)(idx >> 10);
  int tok = inp[sb];
  h[idx] = emb[(size_t)tok * DM_ + d];   // emb_table[0] already zeroed by setup
}

__global__ void posemb_k(float* __restrict__ pe) {
  size_t idx = (size_t)blockIdx.x * blockDim.x + threadIdx.x;
  if (idx >= (size_t)T_ * DM_) return;
  int d = (int)(idx & (DM_ - 1));
  int t = (int)(idx >> 10);
  float pos = (float)(T_ - 1 - t);
  int i = (d < DM_ / 2) ? d : d - DM_ / 2;
  float invf = expf(-9.210340371976184f * (2.0f * (float)i / (float)DM_));
  float ang = pos * invf;
  pe[idx] = (d < DM_ / 2) ? sinf(ang) : cosf(ang);
}

__global__ void concat_k(const float* __restrict__ mem_l, const float* __restrict__ h,
                         float* __restrict__ c) {
  size_t idx = (size_t)blockIdx.x * blockDim.x + threadIdx.x;
  if (idx >= (size_t)T_ * B_ * DM_) return;
  size_t msz = (size_t)MEM_ * B_ * DM_;
  c[idx] = (idx < msz) ? mem_l[idx] : h[idx - msz];
}

// ---------------- f32-WMMA GEMM, async-LDS staged, double buffered ----------------
// C[M,N] = A[M,K] @ B[K,N] (+bias) (+resid) (relu). M mult 16, N mult 64, K mult 16.
// Block = 128 threads (4 waves) -> 16x64 output tile; K staged 16 at a time.
__global__ __launch_bounds__(128) void gemm_k(
    const float* __restrict__ A, const float* __restrict__ Bm, float* __restrict__ C,
    int M, int N, int K,
    const float* __restrict__ bias, const float* __restrict__ resid, int relu) {
  __shared__ float At[2][16][20];   // 16 M x 16 K (+pad)
  __shared__ float Bt[2][16][68];   // 16 K x 64 N (+pad)

  const int t    = threadIdx.x;
  const int lane = t & 31, wv = t >> 5, half = lane >> 4, lid = lane & 15;
  const int nblk = blockIdx.x * 64;
  const int m0   = blockIdx.y * 16;
  const int n0   = nblk + wv * 16;

  // cooperative staging addresses
  const int brow = t >> 3;            // 0..15 (K row of B tile)
  const int bcol = (t & 7) * 8;       // 8 floats = 2 x b128 per thread
  const int arow = (t >> 2) & 15;     // 0..15 (M row of A tile), threads 0..63
  const int acol = (t & 3) * 4;       // 4 floats = 1 x b128

  const float* gB = Bm + (size_t)brow * N + nblk + bcol;
  const float* gA = A + (size_t)(m0 + arow) * K + acol;
  const unsigned lB0 = (unsigned)(uintptr_t)&Bt[0][brow][bcol];
  const unsigned lA0 = (unsigned)(uintptr_t)&At[0][arow][acol];
  const unsigned bStep = (unsigned)(sizeof(float) * 16 * 68);
  const unsigned aStep = (unsigned)(sizeof(float) * 16 * 20);

  const int nb = K >> 4;

  // prologue: stage K-block 0 into buffer 0
  async_b128(lB0, gB);
  async_b128(lB0 + 16, gB + 4);
  if (t < 64) async_b128(lA0, gA);

  v8f acc = {};
  for (int kb = 0; kb < nb; ++kb) {
    wait_async0();       // my wave's async copies for buffer (kb&1) done
    __syncthreads();     // everyone's done; also fences reads of the other buffer
    const int cur = kb & 1;
    if (kb + 1 < nb) {   // stream next K-block into the other buffer
      const int nxt = cur ^ 1;
      const float* gB1 = gB + (size_t)(kb + 1) * 16 * N;
      const float* gA1 = gA + (kb + 1) * 16;
      async_b128(lB0 + nxt * bStep, gB1);
      async_b128(lB0 + nxt * bStep + 16, gB1 + 4);
      if (t < 64) async_b128(lA0 + nxt * aStep, gA1);
    }
#pragma unroll
    for (int kk = 0; kk < 4; ++kk) {
      int k2 = 4 * kk + 2 * half;
      v2f a; a.x = At[cur][lid][k2];          a.y = At[cur][lid][k2 + 1];
      v2f b; b.x = Bt[cur][k2][wv * 16 + lid]; b.y = Bt[cur][k2 + 1][wv * 16 + lid];
      acc = wmma4(a, b, acc);
    }
  }

#pragma unroll
  for (int r = 0; r < 8; ++r) {
    int m = m0 + r + 8 * half;
    int n = n0 + lid;
    float v = acc[r];
    if (bias)  v += bias[n];
    if (resid) v += resid[(size_t)m * N + n];
    if (relu)  v = fmaxf(v, 0.0f);
    C[(size_t)m * N + n] = v;
  }
}

// ---------------- LayerNorm ----------------
__global__ __launch_bounds__(256) void ln_k(const float* __restrict__ x,
                                            const float* __restrict__ g,
                                            const float* __restrict__ bt,
                                            float* __restrict__ out) {
  int row = blockIdx.x;
  const float* xr = x + (size_t)row * DM_;
  float s = 0.0f, ss = 0.0f;
  for (int d = threadIdx.x; d < DM_; d += 256) { float v = xr[d]; s += v; ss += v * v; }
#pragma unroll
  for (int off = 1; off < 32; off <<= 1) { s += __shfl_xor(s, off); ss += __shfl_xor(ss, off); }
  __shared__ float rs[8], rss[8];
  int wv = threadIdx.x >> 5, lane = threadIdx.x & 31;
  if (lane == 0) { rs[wv] = s; rss[wv] = ss; }
  __syncthreads();
  if (wv == 0) {
    s  = (lane < 8) ? rs[lane]  : 0.0f;
    ss = (lane < 8) ? rss[lane] : 0.0f;
#pragma unroll
    for (int off = 1; off < 8; off <<= 1) { s += __shfl_xor(s, off); ss += __shfl_xor(ss, off); }
    if (lane == 0) { rs[0] = s; rss[0] = ss; }
  }
  __syncthreads();
  s = rs[0]; ss = rss[0];
  float mean = s * (1.0f / DM_);
  float var  = ss * (1.0f / DM_) - mean * mean;
  float inv  = rsqrtf(var + 1e-5f);
  float* outr = out + (size_t)row * DM_;
  for (int d = threadIdx.x; d < DM_; d += 256)
    outr[d] = (xr[d] - mean) * inv * g[d] + bt[d];
}

// ---------------- fused Transformer-XL attention (flash-style, all-WMMA) ----------------
// One wave handles one (batch, head, 16-query-row) tile; online softmax over T=2048.
// rel_shift folded in: score[i,j] = (AC[i,j] + BDraw[i, j+S-1-i]) * scale, masked j>i+MEM.
__global__ __launch_bounds__(128) void attn_k(const float* __restrict__ qkv,
                                              const float* __restrict__ rproj,
                                              const float* __restrict__ pbu,
                                              const float* __restrict__ pbv,
                                              float* __restrict__ attn) {
  __shared__ float sP[4][16][17];    // P tile scratch (C/D layout -> A layout)
  __shared__ float sBD[4][16][33];   // sliding 16x32 BDraw window

  int lane = threadIdx.x & 31;
  int wv   = threadIdx.x >> 5;
  int half = lane >> 4;
  int lid  = lane & 15;
  int it    = blockIdx.x * 4 + wv;   // query tile
  int head  = blockIdx.y;
  int batch = blockIdx.z;
  int i0 = it << 4;
  const float scale = 0.125f;        // 1/sqrt(DH)

  // Preload q+u and q+v fragments for all 16 K-steps (K = DH = 64)
  v2f quf[16], qvf[16];
  {
    const float* qrow = qkv + ((size_t)(MEM_ + i0 + lid) * B_ + batch) * (3 * DM_) + head * DH_;
    const float* u  = pbu + head * DH_;
    const float* vb = pbv + head * DH_;
#pragma unroll
    for (int kk = 0; kk < 16; ++kk) {
      int d = 4 * kk + 2 * half;
      float q0 = qrow[d], q1 = qrow[d + 1];
      v2f a; a.x = q0 + u[d];  a.y = q1 + u[d + 1];  quf[kk] = a;
      v2f b; b.x = q0 + vb[d]; b.y = q1 + vb[d + 1]; qvf[kk] = b;
    }
  }

  v8f O0 = {}, O1 = {}, O2 = {}, O3 = {};
  float rowM[8], rowS[8];
#pragma unroll
  for (int r = 0; r < 8; ++r) { rowM[r] = -INFINITY; rowS[r] = 0.0f; }

  // BDraw tile: (q+v) @ r^T for one aligned 16-column jj-tile
  auto bdTile = [&](int jjt) -> v8f {
    v8f acc = {};
    if (jjt < T_) {                  // wave-uniform
      const float* rb = rproj + (size_t)(jjt + lid) * DM_ + head * DH_ + 2 * half;
#pragma unroll
      for (int kk = 0; kk < 16; ++kk) {
        v2f b; b.x = rb[4 * kk]; b.y = rb[4 * kk + 1];
        acc = wmma4(qvf[kk], b, acc);
      }
    }
    return acc;
  };

  int ntiles = (i0 + MEM_) / 16 + 1;        // j-tiles covering unmasked region
  v8f bdA = bdTile((S_ - 16) - i0);         // BDraw tile at base(j0=0)

  for (int jt = 0; jt < ntiles; ++jt) {
    int j0 = jt << 4;
    int base = j0 + (S_ - 16) - i0;         // 16-aligned; window [base, base+32)
    v8f bdB = bdTile(base + 16);

    // AC tile: (q+u) @ k^T
    v8f ac = {};
    {
      const float* kb = qkv + ((size_t)(j0 + lid) * B_ + batch) * (3 * DM_) + DM_ + head * DH_ + 2 * half;
#pragma unroll
      for (int kk = 0; kk < 16; ++kk) {
        v2f b; b.x = kb[4 * kk]; b.y = kb[4 * kk + 1];
        ac = wmma4(quf[kk], b, ac);
      }
    }

    // stage BD window into LDS
#pragma unroll
    for (int r = 0; r < 8; ++r) {
      int m = r + 8 * half;
      sBD[wv][m][lid]      = bdA[r];
      sBD[wv][m][16 + lid] = bdB[r];
    }
    lds_fence();

    // combine + mask + online softmax; write P to LDS
#pragma unroll
    for (int r = 0; r < 8; ++r) {
      int m = r + 8 * half;
      int i = i0 + m;
      int j = j0 + lid;
      float sc = (ac[r] + sBD[wv][m][15 - m + lid]) * scale;  // rel_shift offset
      sc = (j > i + MEM_) ? -1e30f : sc;
      float tm = sc;
#pragma unroll
      for (int off = 1; off < 16; off <<= 1) tm = fmaxf(tm, __shfl_xor(tm, off));
      float Mn = fmaxf(rowM[r], tm);
      float cf = expf(rowM[r] - Mn);
      float p  = expf(sc - Mn);
      float ts = p;
#pragma unroll
      for (int off = 1; off < 16; off <<= 1) ts += __shfl_xor(ts, off);
      rowS[r] = rowS[r] * cf + ts;
      rowM[r] = Mn;
      O0[r] *= cf; O1[r] *= cf; O2[r] *= cf; O3[r] *= cf;
      sP[wv][m][lid] = p;
    }
    lds_fence();

    // PV: O[16x64] += P[16x16] @ V[16x64]
    v2f pa[4];
#pragma unroll
    for (int kk = 0; kk < 4; ++kk) {
      pa[kk].x = sP[wv][lid][4 * kk + 2 * half];
      pa[kk].y = sP[wv][lid][4 * kk + 2 * half + 1];
    }
    const float* vbase = qkv + ((size_t)j0 * B_ + batch) * (3 * DM_) + 2 * DM_ + head * DH_ + lid;
#pragma unroll
    for (int kk = 0; kk < 4; ++kk) {
      int k2 = 4 * kk + 2 * half;
      const float* va = vbase + (size_t)k2 * B_ * (3 * DM_);
      const float* vb = vbase + (size_t)(k2 + 1) * B_ * (3 * DM_);
      v2f b;
      b.x = va[0];  b.y = vb[0];  O0 = wmma4(pa[kk], b, O0);
      b.x = va[16]; b.y = vb[16]; O1 = wmma4(pa[kk], b, O1);
      b.x = va[32]; b.y = vb[32]; O2 = wmma4(pa[kk], b, O2);
      b.x = va[48]; b.y = vb[48]; O3 = wmma4(pa[kk], b, O3);
    }
    bdA = bdB;   // slide window
  }

  // normalize + write out [S, B, NH*DH]
  float* out = attn + head * DH_ + lid;
#pragma unroll
  for (int r = 0; r < 8; ++r) {
    int m = r + 8 * half;
    size_t row = ((size_t)(i0 + m) * B_ + batch) * DM_;
    float inv = 1.0f / rowS[r];
    out[row + 0]  = O0[r] * inv;
    out[row + 16] = O1[r] * inv;
    out[row + 32] = O2[r] * inv;
    out[row + 48] = O3[r] * inv;
  }
}

// ---------------- host orchestration ----------------

extern "C" void kernel_launch(void* const* d_in, const int* in_sizes, int n_in,
                              void* d_out, int out_size, void* d_ws, size_t ws_size,
                              hipStream_t stream) {
  (void)in_sizes; (void)n_in; (void)out_size; (void)ws_size;
  const int*   inputs = (const int*)d_in[0];
  const float* memory = (const float*)d_in[1];
  const float* emb    = (const float*)d_in[2];
  const float* pbu    = (const float*)d_in[3];
  const float* pbv    = (const float*)d_in[4];
  const float* Wqkv   = (const float*)d_in[5];
  const float* Wrel   = (const float*)d_in[6];
  const float* Wo     = (const float*)d_in[7];
  const float* ln1g   = (const float*)d_in[8];
  const float* ln1b   = (const float*)d_in[9];
  const float* Wff1   = (const float*)d_in[10];
  const float* bff1   = (const float*)d_in[11];
  const float* Wff2   = (const float*)d_in[12];
  const float* bff2   = (const float*)d_in[13];
  const float* ln2g   = (const float*)d_in[14];
  const float* ln2b   = (const float*)d_in[15];
  const float* Wdec   = (const float*)d_in[16];

  float* w = (float*)d_ws;                                   // ~160 MB total
  float* posEmb = w; w += (size_t)T_ * DM_;
  float* h      = w; w += (size_t)S_ * B_ * DM_;
  float* cbuf   = w; w += (size_t)T_ * B_ * DM_;
  float* qkvb   = w; w += (size_t)T_ * B_ * 3 * DM_;
  float* rproj  = w; w += (size_t)T_ * DM_;
  float* attnb  = w; w += (size_t)S_ * B_ * DM_;
  float* t1     = w; w += (size_t)S_ * B_ * DM_;
  float* h1     = w; w += (size_t)S_ * B_ * DM_;
  float* ffa    = w; w += (size_t)S_ * B_ * DFF_;
  float* t2     = w; w += (size_t)S_ * B_ * DM_;

  const int MB = S_ * B_;      // 2048 rows for post-attn GEMMs
  const int MT = T_ * B_;      // 4096 rows for QKV GEMM

  {
    size_t tot = (size_t)S_ * B_ * DM_;
    embed_k<<<dim3((unsigned)((tot + 255) / 256)), 256, 0, stream>>>(inputs, emb, h);
  }
  {
    size_t tot = (size_t)T_ * DM_;
    posemb_k<<<dim3((unsigned)((tot + 255) / 256)), 256, 0, stream>>>(posEmb);
  }

  for (int l = 0; l < L_; ++l) {
    {
      size_t tot = (size_t)T_ * B_ * DM_;
      concat_k<<<dim3((unsigned)((tot + 255) / 256)), 256, 0, stream>>>(
          memory + (size_t)l * MEM_ * B_ * DM_, h, cbuf);
    }
    // qkv = c @ W_qkv[l] : [4096,1024]x[1024,3072]
    gemm_k<<<dim3(3 * DM_ / 64, MT / 16), 128, 0, stream>>>(
        cbuf, Wqkv + (size_t)l * DM_ * 3 * DM_, qkvb, MT, 3 * DM_, DM_,
        nullptr, nullptr, 0);
    // rproj = pos_emb @ W_rel[l] : [2048,1024]x[1024,1024]
    gemm_k<<<dim3(DM_ / 64, T_ / 16), 128, 0, stream>>>(
        posEmb, Wrel + (size_t)l * DM_ * DM_, rproj, T_, DM_, DM_,
        nullptr, nullptr, 0);
    // fused rel-attention
    attn_k<<<dim3(S_ / 64, NH_, B_), 128, 0, stream>>>(qkvb, rproj, pbu, pbv, attnb);
    // attn_out = attn @ W_o[l] + h  (residual)
    gemm_k<<<dim3(DM_ / 64, MB / 16), 128, 0, stream>>>(
        attnb, Wo + (size_t)l * DM_ * DM_, t1, MB, DM_, DM_,
        nullptr, h, 0);
    ln_k<<<dim3(MB), 256, 0, stream>>>(t1, ln1g + (size_t)l * DM_, ln1b + (size_t)l * DM_, h1);
    // ff = relu(h1 @ W_ff1 + b1)
    gemm_k<<<dim3(DFF_ / 64, MB / 16), 128, 0, stream>>>(
        h1, Wff1 + (size_t)l * DM_ * DFF_, ffa, MB, DFF_, DM_,
        bff1 + (size_t)l * DFF_, nullptr, 1);
    // t2 = ff @ W_ff2 + b2 + h1
    gemm_k<<<dim3(DM_ / 64, MB / 16), 128, 0, stream>>>(
        ffa, Wff2 + (size_t)l * DFF_ * DM_, t2, MB, DM_, DFF_,
        bff2 + (size_t)l * DM_, h1, 0);
    ln_k<<<dim3(MB), 256, 0, stream>>>(t2, ln2g + (size_t)l * DM_, ln2b + (size_t)l * DM_, h);
  }

  // logits = h @ W_dec : [2048,1024]x[1024,32000]
  gemm_k<<<dim3(V_ / 64, MB / 16), 128, 0, stream>>>(
      h, Wdec, (float*)d_out, MB, V_, DM_, nullptr, nullptr, 0);
}